// VO_50483045597748
// MI455X (gfx1250) — compile-verified
//
#include <hip/hip_runtime.h>

typedef __attribute__((ext_vector_type(2))) float v2f;
typedef __attribute__((ext_vector_type(8))) float v8f;
typedef __attribute__((ext_vector_type(4))) int   v4i;

#define WMMA_F32(a, b, c) \
    __builtin_amdgcn_wmma_f32_16x16x4_f32(false, (a), false, (b), (short)0, (c), false, false)

#ifndef __has_builtin
#define __has_builtin(x) 0
#endif

#if __has_builtin(__builtin_amdgcn_global_load_async_to_lds_b128)
#define HAVE_ASYNC_B128 1
typedef __attribute__((address_space(1))) v4i GASv4i;   // global-AS 16B vector
typedef __attribute__((address_space(3))) v4i LASv4i;   // LDS-AS 16B vector
#else
#define HAVE_ASYNC_B128 0
#endif

#if __has_builtin(__builtin_amdgcn_s_wait_asynccnt)
#define WAIT_ASYNC(n) __builtin_amdgcn_s_wait_asynccnt(n)
#else
#define WAIT_ASYNC(n) asm volatile("s_wait_asynccnt " #n ::: "memory")
#endif

// Problem constants
#define BB 8
#define TT 32
#define NN 10
#define D_OBJ 2048
#define N_VERB 157
#define N_OBJCLS 353
#define D_EMB 512
#define D_VERB 2048
#define M_ROWS (BB * TT * NN)   // 2560
#define K2 (TT * NN)            // 320 (per-batch K for GEMM2)

#define A_STRIDE 36             // 144 B: 16B-aligned vec4 slots, conflict-free frag reads
#define B_STRIDE 68             // 272 B: 16B-aligned vec4 slots, conflict-free frag reads

// ---------------------------------------------------------------------------
// Kernel 1: cls[b,t,n] = last nonzero index of obj_id row (0 if all-zero).
// One wave32 per row; lanes scan strided (coalesced), then max-reduce.
// ---------------------------------------------------------------------------
__global__ __launch_bounds__(256) void vo_cls_kernel(const int* __restrict__ obj_id,
                                                     int* __restrict__ cls) {
    const int row  = blockIdx.x * 8 + (threadIdx.x >> 5);
    const int lane = threadIdx.x & 31;
    if (row >= M_ROWS) return;
    const int* p = obj_id + (long)row * N_OBJCLS;
    int best = -1;
    for (int j = lane; j < N_OBJCLS; j += 32)
        if (p[j] != 0) best = j;
    #pragma unroll
    for (int off = 16; off > 0; off >>= 1) {
        int o = __shfl_xor(best, off, 32);
        best = best > o ? best : o;
    }
    if (lane == 0) cls[row] = (best < 0) ? 0 : best;
}

// ---------------------------------------------------------------------------
// Kernel 2: E[m, :] = feat[m, :] @ W + b   (M=2560, K=2048, N=512), f32 WMMA.
// WG tile 128x64, 8 waves, each wave 32x32 (2x2 fragments of 16x16).
// Double-buffered LDS, staged via async global->LDS B128 (ASYNCcnt pipelined).
// ---------------------------------------------------------------------------
__global__ __launch_bounds__(256) void vo_gemm1_kernel(const float* __restrict__ feat,
                                                       const float* __restrict__ W,
                                                       const float* __restrict__ bias,
                                                       float* __restrict__ E) {
    __shared__ float As[2][128 * A_STRIDE];  // As[m][k], row-major, stride 36
    __shared__ float Bs[2][32 * B_STRIDE];   // Bs[k][n], row-major, stride 68

    const int m0   = blockIdx.x * 128;
    const int n0   = blockIdx.y * 64;
    const int tid  = threadIdx.x;
    const int lane = tid & 31, wave = tid >> 5;
    const int wm   = (wave & 3) * 32;      // wave M offset within 128
    const int wn   = (wave >> 2) * 32;     // wave N offset within 64
    const int half = lane >> 4, lrow = lane & 15;

    // Stage one 32-wide K chunk: 4 + 2 = 6 B128 transfers per lane.
    auto stage = [&](int buf, int kc) {
        #pragma unroll
        for (int i = 0; i < 4; ++i) {               // A: 128 x 32 = 1024 vec4 slots
            int vi = tid + i * 256;
            int m  = vi >> 3;
            int kq = (vi & 7) * 4;
            const float* g = feat + (long)(m0 + m) * D_OBJ + kc + kq;
            float*       l = &As[buf][m * A_STRIDE + kq];
#if HAVE_ASYNC_B128
            __builtin_amdgcn_global_load_async_to_lds_b128((GASv4i*)g, (LASv4i*)l, 0, 0);
#else
            *(float4*)l = *(const float4*)g;
#endif
        }
        #pragma unroll
        for (int i = 0; i < 2; ++i) {               // B: 32 x 64 = 512 vec4 slots
            int vi = tid + i * 256;
            int k  = vi >> 4;
            int nq = (vi & 15) * 4;
            const float* g = W + (long)(kc + k) * D_EMB + n0 + nq;
            float*       l = &Bs[buf][k * B_STRIDE + nq];
#if HAVE_ASYNC_B128
            __builtin_amdgcn_global_load_async_to_lds_b128((GASv4i*)g, (LASv4i*)l, 0, 0);
#else
            *(float4*)l = *(const float4*)g;
#endif
        }
    };

    v8f acc[2][2] = {};
    stage(0, 0);
    int buf = 0;

    for (int kc = 0; kc < D_OBJ; kc += 32) {
        const bool more = (kc + 32) < D_OBJ;
        if (more) stage(buf ^ 1, kc + 32);          // prefetch next chunk (async)
#if HAVE_ASYNC_B128
        if (more) WAIT_ASYNC(6);                    // current chunk done; 6 prefetch ops in flight
        else      WAIT_ASYNC(0);
#endif
        __syncthreads();

        const float* A_ = As[buf];
        const float* B_ = Bs[buf];
        #pragma unroll
        for (int kk = 0; kk < 32; kk += 4) {
            const int kb = kk + 2 * half;           // lanes 0-15 -> K=kk,kk+1; 16-31 -> kk+2,kk+3
            v2f af[2], bf[2];
            #pragma unroll
            for (int mi = 0; mi < 2; ++mi) {
                const float* ar = A_ + (wm + mi * 16 + lrow) * A_STRIDE + kb;
                af[mi].x = ar[0];
                af[mi].y = ar[1];
            }
            #pragma unroll
            for (int ni = 0; ni < 2; ++ni) {
                bf[ni].x = B_[kb * B_STRIDE + wn + ni * 16 + lrow];
                bf[ni].y = B_[(kb + 1) * B_STRIDE + wn + ni * 16 + lrow];
            }
            #pragma unroll
            for (int mi = 0; mi < 2; ++mi)
                #pragma unroll
                for (int ni = 0; ni < 2; ++ni)
                    acc[mi][ni] = WMMA_F32(af[mi], bf[ni], acc[mi][ni]);
        }
        __syncthreads();
        buf ^= 1;
    }

    // Epilogue: fuse bias, store E. C/D layout: vgpr i -> row (i + 8*half), col lrow.
    #pragma unroll
    for (int mi = 0; mi < 2; ++mi)
        #pragma unroll
        for (int ni = 0; ni < 2; ++ni) {
            const int n  = n0 + wn + ni * 16 + lrow;
            const float bv = bias[n];
            #pragma unroll
            for (int i = 0; i < 8; ++i) {
                const int m = m0 + wm + mi * 16 + 8 * half + i;
                E[(long)m * D_EMB + n] = acc[mi][ni][i] + bv;
            }
        }
}

// ---------------------------------------------------------------------------
// Kernel 3: out1[b,v,d] = (1/T) * sum_m A_o2v[v, cls[b,m]] * E[b,m,d]
// M=157 (padded 192), N=512, K=320. A-frags gathered from L2-resident A_o2v,
// B-frags read from E; cls_b staged in LDS.
// ---------------------------------------------------------------------------
__global__ __launch_bounds__(256) void vo_gemm2_kernel(const float* __restrict__ A_o2v,
                                                       const float* __restrict__ E,
                                                       const int* __restrict__ cls,
                                                       float* __restrict__ out) {
    __shared__ int clsS[K2];
    const int b    = blockIdx.z;
    const int n0   = blockIdx.y * 64;
    const int tid  = threadIdx.x;
    const int lane = tid & 31, wave = tid >> 5;
    const int wm   = (wave & 3) * 16;      // 4 M-tiles of 16 within 64-row chunk
    const int wn   = (wave >> 2) * 32;     // 2 N-halves of 32
    const int half = lane >> 4, lrow = lane & 15;

    for (int idx = tid; idx < K2; idx += 256) clsS[idx] = cls[b * K2 + idx];
    __syncthreads();

    const int vbase = blockIdx.x * 64 + wm;
    int vrow = vbase + lrow;
    if (vrow > N_VERB - 1) vrow = N_VERB - 1;          // clamp loads; stores masked below
    const float* Arow = A_o2v + (long)vrow * N_OBJCLS;
    const float* Eb   = E + (long)b * K2 * D_EMB + n0 + wn;

    v8f acc[2] = {};
    #pragma unroll 4
    for (int ks = 0; ks < K2; ks += 4) {
        const int kb = ks + 2 * half;
        v2f af, bf0, bf1;
        af.x = Arow[clsS[kb]];
        af.y = Arow[clsS[kb + 1]];
        const float* e0 = Eb + (long)kb * D_EMB;
        bf0.x = e0[lrow];           bf0.y = e0[D_EMB + lrow];
        bf1.x = e0[16 + lrow];      bf1.y = e0[D_EMB + 16 + lrow];
        acc[0] = WMMA_F32(af, bf0, acc[0]);
        acc[1] = WMMA_F32(af, bf1, acc[1]);
    }

    #pragma unroll
    for (int ni = 0; ni < 2; ++ni) {
        const int d = n0 + wn + ni * 16 + lrow;
        #pragma unroll
        for (int i = 0; i < 8; ++i) {
            const int v = vbase + 8 * half + i;
            if (v < N_VERB)
                out[((long)b * N_VERB + v) * (D_EMB + D_VERB) + d] =
                    acc[ni][i] * (1.0f / (float)TT);
        }
    }
}

// ---------------------------------------------------------------------------
// Kernel 4: out[b, v, 512 + d] = mean_t fm_context[b, t, d]   for all v.
// ---------------------------------------------------------------------------
__global__ __launch_bounds__(256) void vo_ctx_kernel(const float* __restrict__ fm,
                                                     float* __restrict__ out) {
    const int idx = blockIdx.x * 256 + threadIdx.x;  // 8 * 2048 threads
    const int b = idx >> 11;
    const int d = idx & (D_VERB - 1);
    const float* p = fm + (long)b * TT * D_VERB + d;
    float s = 0.f;
    #pragma unroll
    for (int t = 0; t < TT; ++t) s += p[(long)t * D_VERB];
    s *= (1.0f / (float)TT);
    float* o = out + (long)b * N_VERB * (D_EMB + D_VERB) + D_EMB + d;
    for (int v = 0; v < N_VERB; ++v) o[(long)v * (D_EMB + D_VERB)] = s;
}

// ---------------------------------------------------------------------------
extern "C" void kernel_launch(void* const* d_in, const int* in_sizes, int n_in,
                              void* d_out, int out_size, void* d_ws, size_t ws_size,
                              hipStream_t stream) {
    const float* feat   = (const float*)d_in[0];  // (8,32,10,2048)
    const float* fm     = (const float*)d_in[1];  // (8,32,2048)
    const int*   obj_id = (const int*)d_in[2];    // (8,32,10,353)
    const float* W      = (const float*)d_in[3];  // (2048,512)
    const float* bias   = (const float*)d_in[4];  // (512,)
    const float* A_o2v  = (const float*)d_in[5];  // (157,353)
    float* out = (float*)d_out;                   // (8,157,2560)

    int*   cls = (int*)d_ws;                            // 2560 ints
    float* E   = (float*)((char*)d_ws + 16384);         // 2560 x 512 floats

    vo_cls_kernel<<<M_ROWS / 8, 256, 0, stream>>>(obj_id, cls);
    vo_gemm1_kernel<<<dim3(M_ROWS / 128, D_EMB / 64), 256, 0, stream>>>(feat, W, bias, E);
    vo_gemm2_kernel<<<dim3(3, D_EMB / 64, BB), 256, 0, stream>>>(A_o2v, E, cls, out);
    vo_ctx_kernel<<<(BB * D_VERB) / 256, 256, 0, stream>>>(fm, out);
}